// EntropicGCN_72627896975950
// MI455X (gfx1250) — compile-verified
//
#include <hip/hip_runtime.h>
#include <math.h>

#define DIN  128
#define DH   128
#define DOUT 64
#define LN_EPS 1e-5f
#define WEIGHT 0.5f

typedef float v2f __attribute__((ext_vector_type(2)));
typedef float v8f __attribute__((ext_vector_type(8)));

// ---------------- wave helpers (wave32) ----------------
__device__ __forceinline__ float waveSum(float v) {
  #pragma unroll
  for (int off = 16; off > 0; off >>= 1) v += __shfl_xor(v, off, 32);
  return v;
}
__device__ __forceinline__ float waveMax(float v) {
  #pragma unroll
  for (int off = 16; off > 0; off >>= 1) v = fmaxf(v, __shfl_xor(v, off, 32));
  return v;
}
__device__ __forceinline__ void atomicMaxFloat(float* addr, float val) {
  if (val >= 0.0f) atomicMax((int*)addr, __float_as_int(val));
  else             atomicMin((unsigned int*)addr, __float_as_uint(val));
}

// ---------------- utility kernels ----------------
__global__ void k_zero4(float4* __restrict__ p, long long n4) {
  long long i = (long long)blockIdx.x * blockDim.x + threadIdx.x;
  if (i < n4) p[i] = make_float4(0.f, 0.f, 0.f, 0.f);
}
__global__ void k_init_scalars(float* s) {
  if (threadIdx.x == 0 && blockIdx.x == 0) {
    s[0] = -INFINITY; // running max of logits
    s[1] = 0.0f;      // sum(exp)
    s[2] = 0.0f;      // entropy S
    s[3] = 0.0f;
  }
}

// ---------------- degree counting ----------------
__global__ void k_count_deg(const long long* __restrict__ ei, long long E,
                            float* __restrict__ cnt_src, float* __restrict__ cnt_dst) {
  long long e = (long long)blockIdx.x * blockDim.x + threadIdx.x;
  if (e < E) {
    atomicAdd(&cnt_src[ei[e]], 1.0f);
    atomicAdd(&cnt_dst[ei[E + e]], 1.0f);
  }
}
// deg_adj holds cnt_src; dinv holds cnt_dst on entry.
__global__ void k_finalize_deg(float* __restrict__ deg_adj, float* __restrict__ dinv,
                               float* __restrict__ invdeg, float* __restrict__ en_norm, int nN) {
  int n = blockIdx.x * blockDim.x + threadIdx.x;
  if (n >= nN) return;
  float cd = dinv[n];                  // cnt over dst
  float deg = cd + 1.0f;               // incl. self-loop
  dinv[n]   = rsqrtf(deg);
  invdeg[n] = 1.0f / deg;
  en_norm[n] = rsqrtf(deg_adj[n] * (float)DH); // 1/sqrt(deg_adj*DH)
}

// ---------------- WMMA GEMM: one wave computes a 16 x NC strip ----------------
// fp32 V_WMMA_F32_16X16X4_F32; A fragment reused across NC/16 column tiles ->
// NC/16 independent accumulator chains per wave (good XDL pipelining).
template <int NC>
__global__ void k_gemm_wmma(const float* __restrict__ A, const float* __restrict__ B,
                            float* __restrict__ C, int M, int K) {
  int wave = (int)(((long long)blockIdx.x * blockDim.x + threadIdx.x) >> 5);
  int lane = threadIdx.x & 31;
  int tilesM = (M + 15) >> 4;
  if (wave >= tilesM) return;          // wave-uniform: EXEC stays all-ones for WMMA
  constexpr int NT = NC / 16;
  int half = lane >> 4;                // 0: lanes 0-15, 1: lanes 16-31
  int mn   = lane & 15;
  int row  = wave * 16 + mn;           // A row for this lane

  v8f acc[NT];
  #pragma unroll
  for (int t = 0; t < NT; ++t) acc[t] = (v8f){};

  for (int k0 = 0; k0 < K; k0 += 4) {
    int ka = k0 + 2 * half;
    v2f a;                             // contiguous pair -> global_load_b64
    a.x = A[(size_t)row * K + ka];
    a.y = A[(size_t)row * K + ka + 1];
    #pragma unroll
    for (int t = 0; t < NT; ++t) {
      int col = t * 16 + mn;
      v2f b;
      b.x = B[(size_t)ka * NC + col];
      b.y = B[(size_t)(ka + 1) * NC + col];
      acc[t] = __builtin_amdgcn_wmma_f32_16x16x4_f32(false, a, false, b, (short)0,
                                                     acc[t], false, false);
    }
  }
  // D layout: VGPR v -> M = wave*16 + v + 8*half, N = t*16 + mn
  #pragma unroll
  for (int t = 0; t < NT; ++t) {
    #pragma unroll
    for (int v = 0; v < 8; ++v) {
      int r = wave * 16 + v + 8 * half;
      if (r < M) C[(size_t)r * NC + t * 16 + mn] = acc[t][v];
    }
  }
}

// ---------------- edge aggregate (C=128): agg[dst] += hw[src]*dinv[s]*dinv[d] ----------------
__global__ void k_edge_agg128(const long long* __restrict__ ei, long long E,
                              const float* __restrict__ hw, const float* __restrict__ dinv,
                              float* __restrict__ agg) {
  long long wid = ((long long)blockIdx.x * blockDim.x + threadIdx.x) >> 5;
  int lane = threadIdx.x & 31;
  if (wid >= E) return;
  long long s = ei[wid], d = ei[E + wid];
  float coeff = dinv[s] * dinv[d];
  const float4 v = *(const float4*)(hw + (size_t)s * DH + lane * 4);   // b128 load
  float* base = agg + (size_t)d * DH + lane * 4;
  atomicAdd(base + 0, v.x * coeff);
  atomicAdd(base + 1, v.y * coeff);
  atomicAdd(base + 2, v.z * coeff);
  atomicAdd(base + 3, v.w * coeff);
}

// ---------------- edge aggregate (C=64) ----------------
__global__ void k_edge_agg64(const long long* __restrict__ ei, long long E,
                             const float* __restrict__ hw, const float* __restrict__ dinv,
                             float* __restrict__ agg) {
  long long wid = ((long long)blockIdx.x * blockDim.x + threadIdx.x) >> 5;
  int lane = threadIdx.x & 31;
  if (wid >= E) return;
  long long s = ei[wid], d = ei[E + wid];
  float coeff = dinv[s] * dinv[d];
  const float2 v = *(const float2*)(hw + (size_t)s * DOUT + lane * 2); // b64 load
  float* base = agg + (size_t)d * DOUT + lane * 2;
  atomicAdd(base + 0, v.x * coeff);
  atomicAdd(base + 1, v.y * coeff);
}

// ---------------- hidden layer epilogue: bias + self + ReLU + LayerNorm ----------------
__global__ void k_node_relu_ln(const float* __restrict__ hw, const float* __restrict__ agg,
                               const float* __restrict__ invdeg, const float* __restrict__ bias,
                               const float* __restrict__ gamma, const float* __restrict__ beta,
                               float* __restrict__ out, int nN) {
  long long wid = ((long long)blockIdx.x * blockDim.x + threadIdx.x) >> 5;
  int lane = threadIdx.x & 31;
  if (wid >= nN) return;
  float idg = invdeg[wid];
  int c0 = lane * 4;
  const float4 va = *(const float4*)(agg + (size_t)wid * DH + c0);
  const float4 vh = *(const float4*)(hw + (size_t)wid * DH + c0);
  const float4 vb = *(const float4*)(bias + c0);
  float vals[4];
  vals[0] = fmaxf(va.x + vh.x * idg + vb.x, 0.f);
  vals[1] = fmaxf(va.y + vh.y * idg + vb.y, 0.f);
  vals[2] = fmaxf(va.z + vh.z * idg + vb.z, 0.f);
  vals[3] = fmaxf(va.w + vh.w * idg + vb.w, 0.f);
  float s = vals[0] + vals[1] + vals[2] + vals[3];
  float mean = waveSum(s) * (1.0f / DH);
  float var = 0.f;
  #pragma unroll
  for (int i = 0; i < 4; ++i) { float d = vals[i] - mean; var += d * d; }
  var = waveSum(var) * (1.0f / DH);
  float r = rsqrtf(var + LN_EPS);
  const float4 vg = *(const float4*)(gamma + c0);
  const float4 vt = *(const float4*)(beta + c0);
  float4 o;
  o.x = (vals[0] - mean) * r * vg.x + vt.x;
  o.y = (vals[1] - mean) * r * vg.y + vt.y;
  o.z = (vals[2] - mean) * r * vg.z + vt.z;
  o.w = (vals[3] - mean) * r * vg.w + vt.w;
  *(float4*)(out + (size_t)wid * DH + c0) = o;
}

// ---------------- output layer epilogue (no ReLU/LN), C = 64 ----------------
__global__ void k_node_bias(const float* __restrict__ hw, const float* __restrict__ agg,
                            const float* __restrict__ invdeg, const float* __restrict__ bias,
                            float* __restrict__ emb, int nN) {
  long long wid = ((long long)blockIdx.x * blockDim.x + threadIdx.x) >> 5;
  int lane = threadIdx.x & 31;
  if (wid >= nN) return;
  float idg = invdeg[wid];
  int c0 = lane * 2;
  const float2 va = *(const float2*)(agg + (size_t)wid * DOUT + c0);
  const float2 vh = *(const float2*)(hw + (size_t)wid * DOUT + c0);
  const float2 vb = *(const float2*)(bias + c0);
  float2 o;
  o.x = va.x + vh.x * idg + vb.x;
  o.y = va.y + vh.y * idg + vb.y;
  *(float2*)(emb + (size_t)wid * DOUT + c0) = o;
}

// ---------------- rn2[n] = ||emb[n]||^2 ----------------
__global__ void k_rn2(const float* __restrict__ emb, float* __restrict__ rn2, int nN) {
  long long wid = ((long long)blockIdx.x * blockDim.x + threadIdx.x) >> 5;
  int lane = threadIdx.x & 31;
  if (wid >= nN) return;
  const float2 v = *(const float2*)(emb + (size_t)wid * DOUT + lane * 2);
  float s = waveSum(v.x * v.x + v.y * v.y);
  if (lane == 0) rn2[wid] = s;
}

// ---------------- entropy edge pass A: dot, res2, res3, nsumX ----------------
__global__ void k_edge_entropy_a(const long long* __restrict__ ei, long long E,
                                 const float* __restrict__ emb, const float* __restrict__ rn2,
                                 float* __restrict__ res2, float* __restrict__ res3,
                                 float* __restrict__ nsumX) {
  long long wid = ((long long)blockIdx.x * blockDim.x + threadIdx.x) >> 5;
  int lane = threadIdx.x & 31;
  if (wid >= E) return;
  long long s = ei[wid], d = ei[E + wid];
  int c0 = lane * 2;
  const float2 es = *(const float2*)(emb + (size_t)s * DOUT + c0);
  const float2 ed = *(const float2*)(emb + (size_t)d * DOUT + c0);
  float* nb = nsumX + (size_t)s * DOUT + c0;
  atomicAdd(nb + 0, ed.x);
  atomicAdd(nb + 1, ed.y);
  float dot = waveSum(es.x * ed.x + es.y * ed.y);
  if (lane == 0) {
    atomicAdd(&res2[s], dot);
    atomicAdd(&res3[s], rn2[d]);
  }
}

// ---------------- logits + running max ----------------
__global__ void k_logits(const float* __restrict__ deg_adj, const float* __restrict__ rn2,
                         const float* __restrict__ res2, const float* __restrict__ res3,
                         const float* __restrict__ en_norm, float* __restrict__ buf,
                         float* __restrict__ scal, int nN) {
  int n = blockIdx.x * blockDim.x + threadIdx.x;
  float lg = -INFINITY;
  if (n < nN) {
    float e = 0.5f * (deg_adj[n] * rn2[n] - 2.0f * res2[n] + res3[n]) * en_norm[n];
    lg = -e;                    // TEMP = 1
    buf[n] = lg;
  }
  float m = waveMax(lg);
  if ((threadIdx.x & 31) == 0) atomicMaxFloat(&scal[0], m);
}

__global__ void k_expsum(float* __restrict__ buf, const float* __restrict__ scal,
                         float* __restrict__ scal_out, int nN) {
  int n = blockIdx.x * blockDim.x + threadIdx.x;
  float e = 0.0f;
  if (n < nN) {
    e = expf(buf[n] - scal[0]);
    buf[n] = e;
  }
  e = waveSum(e);
  if ((threadIdx.x & 31) == 0) atomicAdd(&scal_out[1], e);
}

__global__ void k_entropy(float* __restrict__ buf, const float* __restrict__ scal,
                          float* __restrict__ scal_out, int nN) {
  int n = blockIdx.x * blockDim.x + threadIdx.x;
  float t = 0.0f;
  if (n < nN) {
    float P = buf[n] / scal[1] + 1e-10f;
    buf[n] = P;
    t = -P * logf(P);
  }
  t = waveSum(t);
  if ((threadIdx.x & 31) == 0) atomicAdd(&scal_out[2], t);
}

__global__ void k_pbar(float* __restrict__ buf, const float* __restrict__ scal, int nN) {
  int n = blockIdx.x * blockDim.x + threadIdx.x;
  if (n < nN) {
    float P = buf[n];
    buf[n] = P * (scal[2] + logf(P));   // Pbar
  }
}

// ---------------- entropy edge pass B: nsumP, g4 ----------------
__global__ void k_edge_entropy_b(const long long* __restrict__ ei, long long E,
                                 const float* __restrict__ emb, const float* __restrict__ pbar,
                                 float* __restrict__ nsumP, float* __restrict__ g4) {
  long long wid = ((long long)blockIdx.x * blockDim.x + threadIdx.x) >> 5;
  int lane = threadIdx.x & 31;
  if (wid >= E) return;
  long long s = ei[wid], d = ei[E + wid];
  float pbd = pbar[d];
  int c0 = lane * 2;
  const float2 ed = *(const float2*)(emb + (size_t)d * DOUT + c0);
  float* gb = g4 + (size_t)s * DOUT + c0;
  atomicAdd(gb + 0, ed.x * pbd);
  atomicAdd(gb + 1, ed.y * pbd);
  if (lane == 0) atomicAdd(&nsumP[s], pbd);
}

// ---------------- final: out = emb + 0.5 * grad_S ----------------
__global__ void k_final(const float* __restrict__ emb, const float* __restrict__ pbar,
                        const float* __restrict__ nsumP, const float* __restrict__ nsumX,
                        const float* __restrict__ g4, const float* __restrict__ deg_adj,
                        const float* __restrict__ en_norm, float* __restrict__ out,
                        long long total) {
  long long i = (long long)blockIdx.x * blockDim.x + threadIdx.x;
  if (i >= total) return;
  int n = (int)(i / DOUT);
  float x  = emb[i];
  float pb = pbar[n];
  float g = (deg_adj[n] * x * pb + x * nsumP[n] - pb * nsumX[i] - g4[i]) * en_norm[n];
  out[i] = x + WEIGHT * g;   // TEMP = 1
}

// =========================== host launcher ===========================
extern "C" void kernel_launch(void* const* d_in, const int* in_sizes, int n_in,
                              void* d_out, int out_size, void* d_ws, size_t ws_size,
                              hipStream_t stream) {
  const float*     x     = (const float*)d_in[0];
  const long long* ei    = (const long long*)d_in[1];
  const float*     W0    = (const float*)d_in[2];
  const float*     b0    = (const float*)d_in[3];
  const float*     W1    = (const float*)d_in[4];
  const float*     b1    = (const float*)d_in[5];
  const float*     W2    = (const float*)d_in[6];
  const float*     b2    = (const float*)d_in[7];
  const float*     Wout  = (const float*)d_in[8];
  const float*     bout  = (const float*)d_in[9];
  const float*     gamma = (const float*)d_in[10];
  const float*     beta  = (const float*)d_in[11];
  float* out = (float*)d_out;

  const int       nN = in_sizes[0] / DIN;       // 50000
  const long long E  = (long long)in_sizes[1] / 2;

  // ---- workspace carve-out (floats; all region sizes are multiples of 4) ----
  float* W = (float*)d_ws;
  size_t off = 0;
  auto alloc = [&](size_t n) { float* p = W + off; off += n; return p; };
  float* deg_adj = alloc(nN);                // cnt_src
  float* dinv    = alloc(nN);                // cnt_dst -> rsqrt(deg)
  float* invdeg  = alloc(nN);
  float* en_norm = alloc(nN);
  float* h_cur   = alloc((size_t)nN * DH);
  float* hw      = alloc((size_t)nN * DH);
  float* agg     = alloc((size_t)nN * DH);
  float* emb     = alloc((size_t)nN * DOUT);
  float* rn2     = alloc(nN);
  float* res2    = alloc(nN);                // res2,res3,buf,nsumP contiguous
  float* res3    = alloc(nN);
  float* buf     = alloc(nN);                // logits -> exp -> P -> Pbar
  float* nsumP   = alloc(nN);
  float* nsumX   = alloc((size_t)nN * DOUT); // nsumX,g4 contiguous
  float* g4      = alloc((size_t)nN * DOUT);
  float* scal    = alloc(4);
  (void)ws_size; (void)n_in; (void)out_size;

  const int TB = 256;
  auto blocksFor = [](long long n, int tb) { return (unsigned)((n + tb - 1) / tb); };
  auto waveBlocks = [](long long waves) { return (unsigned)((waves * 32 + 255) / 256); };
  auto zero = [&](float* p, long long n) {
    k_zero4<<<blocksFor(n / 4, TB), TB, 0, stream>>>((float4*)p, n / 4);
  };

  // ---- degrees ----
  zero(deg_adj, 2LL * nN);
  k_count_deg<<<blocksFor(E, TB), TB, 0, stream>>>(ei, E, deg_adj, dinv);
  k_finalize_deg<<<blocksFor(nN, TB), TB, 0, stream>>>(deg_adj, dinv, invdeg, en_norm, nN);

  // ---- 3 GCN layers (fp32 WMMA GEMM + edge scatter + fused epilogue) ----
  const float* Ws[3] = {W0, W1, W2};
  const float* bs[3] = {b0, b1, b2};
  const long long tilesM = (nN + 15) / 16;
  for (int l = 0; l < 3; ++l) {
    const float* A = (l == 0) ? x : h_cur;
    k_gemm_wmma<DH><<<waveBlocks(tilesM), TB, 0, stream>>>(A, Ws[l], hw, nN, DH);
    zero(agg, (long long)nN * DH);
    k_edge_agg128<<<waveBlocks(E), TB, 0, stream>>>(ei, E, hw, dinv, agg);
    k_node_relu_ln<<<waveBlocks(nN), TB, 0, stream>>>(hw, agg, invdeg, bs[l], gamma, beta,
                                                      h_cur, nN);
  }

  // ---- output layer (DOUT = 64) ----
  k_gemm_wmma<DOUT><<<waveBlocks(tilesM), TB, 0, stream>>>(h_cur, Wout, hw, nN, DH);
  zero(agg, (long long)nN * DOUT);
  k_edge_agg64<<<waveBlocks(E), TB, 0, stream>>>(ei, E, hw, dinv, agg);
  k_node_bias<<<waveBlocks(nN), TB, 0, stream>>>(hw, agg, invdeg, bout, emb, nN);

  // ---- entropy gradient ----
  k_rn2<<<waveBlocks(nN), TB, 0, stream>>>(emb, rn2, nN);
  zero(res2, 4LL * nN);                       // res2, res3, buf, nsumP
  zero(nsumX, 2LL * nN * DOUT);               // nsumX + g4
  k_init_scalars<<<1, 1, 0, stream>>>(scal);
  k_edge_entropy_a<<<waveBlocks(E), TB, 0, stream>>>(ei, E, emb, rn2, res2, res3, nsumX);
  k_logits<<<blocksFor(nN, TB), TB, 0, stream>>>(deg_adj, rn2, res2, res3, en_norm, buf,
                                                 scal, nN);
  k_expsum<<<blocksFor(nN, TB), TB, 0, stream>>>(buf, scal, scal, nN);
  k_entropy<<<blocksFor(nN, TB), TB, 0, stream>>>(buf, scal, scal, nN);
  k_pbar<<<blocksFor(nN, TB), TB, 0, stream>>>(buf, scal, nN);
  k_edge_entropy_b<<<waveBlocks(E), TB, 0, stream>>>(ei, E, emb, buf, nsumP, g4);
  k_final<<<blocksFor((long long)nN * DOUT, TB), TB, 0, stream>>>(
      emb, buf, nsumP, nsumX, g4, deg_adj, en_norm, out, (long long)nN * DOUT);
}